// LightGCN2_12575664242811
// MI455X (gfx1250) — compile-verified
//
#include <hip/hip_runtime.h>

// ---------------------------------------------------------------------------
// LightGCN 3-layer COO SpMM for MI455X (gfx1250).
// Memory/atomic-bound (~0.25 FLOP/B): everything lives in the 192MB L2.
// CDNA5 paths used: GLOBAL_LOAD_ASYNC_TO_LDS_B32 (ASYNCcnt) for edge staging,
// global_prefetch_b8 for gather rows, global_atomic_add_f32 scatter.
// ---------------------------------------------------------------------------

#ifndef __has_builtin
#define __has_builtin(x) 0
#endif

#if __has_builtin(__builtin_amdgcn_global_load_async_to_lds_b32) && \
    __has_builtin(__builtin_amdgcn_s_wait_asynccnt)
#define LGCN_ASYNC 1
#else
#define LGCN_ASYNC 0
#endif

// Cast helpers for the async-to-LDS builtin: (global int*, AS3 int*, imm, imm)
#define LGCN_GPTR(p) ((int*)(void*)(p))
#define LGCN_LPTR(p) \
    ((__attribute__((address_space(3))) int*)(__attribute__((address_space(3))) void*)(p))

#define LGCN_TILE 256   // edges staged per block
#define LGCN_LPE  16    // lanes cooperating on one edge (64 floats / float4)

__global__ __launch_bounds__(256) void lgcn_scatter(
    const int* __restrict__ rows, const int* __restrict__ cols,
    const float* __restrict__ vals, const float* __restrict__ x,
    float* __restrict__ y, int nEdges)
{
    __shared__ int   sRow[LGCN_TILE];
    __shared__ int   sCol[LGCN_TILE];
    __shared__ float sVal[LGCN_TILE];

    const int t = threadIdx.x;
    const long long base = (long long)blockIdx.x * LGCN_TILE;
    const long long eg0  = base + t;

#if LGCN_ASYNC
    // Async DMA of this block's edge triples into LDS (tracked on ASYNCcnt).
    if (eg0 < nEdges) {
        __builtin_amdgcn_global_load_async_to_lds_b32(
            LGCN_GPTR(rows + eg0), LGCN_LPTR(&sRow[t]), 0, 0);
        __builtin_amdgcn_global_load_async_to_lds_b32(
            LGCN_GPTR(cols + eg0), LGCN_LPTR(&sCol[t]), 0, 0);
        __builtin_amdgcn_global_load_async_to_lds_b32(
            LGCN_GPTR(vals + eg0), LGCN_LPTR(&sVal[t]), 0, 0);
    }
    __builtin_amdgcn_s_wait_asynccnt(0);
    __syncthreads();
#else
    if (eg0 < nEdges) {
        sRow[t] = rows[eg0];
        sCol[t] = cols[eg0];
        sVal[t] = vals[eg0];
    }
    __syncthreads();
#endif

    const int sub = t & (LGCN_LPE - 1);   // lane within edge group (0..15)
    const int grp = t >> 4;               // edge subgroup (0..15)

    #pragma unroll 4
    for (int k = 0; k < LGCN_TILE / LGCN_LPE; ++k) {
        const int el = (grp << 4) + k;
        const long long eg = base + el;
        if (eg >= nEdges) break;          // eg increases with k per subgroup

        const int   r = sRow[el];
        const int   c = sCol[el];
        const float v = sVal[el];

        // Prefetch next edge's gather row (global_prefetch_b8).
        if (k + 1 < LGCN_TILE / LGCN_LPE && eg + 1 < nEdges) {
            const int cn = sCol[el + 1];
            __builtin_prefetch(x + (((long long)cn) << 6) + (sub << 2), 0, 0);
        }

        // Coalesced 256B row gather: 16 lanes x float4 (global_load_b128).
        const float4 xv =
            *(const float4*)(x + (((long long)c) << 6) + (sub << 2));

        // Scatter-add into 4 contiguous f32 (global_atomic_add_f32, no return).
        float* yp = y + (((long long)r) << 6) + (sub << 2);
        atomicAdd(yp + 0, v * xv.x);
        atomicAdd(yp + 1, v * xv.y);
        atomicAdd(yp + 2, v * xv.z);
        atomicAdd(yp + 3, v * xv.w);
    }
}

// out = u ; z = 0   (vectorized, one pass)
__global__ __launch_bounds__(256) void lgcn_init(
    float4* __restrict__ out, const float4* __restrict__ u,
    float4* __restrict__ z, int n4)
{
    const int i = blockIdx.x * blockDim.x + threadIdx.x;
    if (i < n4) {
        out[i] = u[i];
        z[i]   = make_float4(0.f, 0.f, 0.f, 0.f);
    }
}

// out += inc ; optionally z = 0 for the next layer's accumulator
__global__ __launch_bounds__(256) void lgcn_add_zero(
    float4* __restrict__ out, const float4* __restrict__ inc,
    float4* __restrict__ z, int n4)
{
    const int i = blockIdx.x * blockDim.x + threadIdx.x;
    if (i < n4) {
        float4 o = out[i];
        const float4 a = inc[i];
        o.x += a.x; o.y += a.y; o.z += a.z; o.w += a.w;
        out[i] = o;
        if (z) z[i] = make_float4(0.f, 0.f, 0.f, 0.f);
    }
}

extern "C" void kernel_launch(void* const* d_in, const int* in_sizes, int n_in,
                              void* d_out, int out_size, void* d_ws, size_t ws_size,
                              hipStream_t stream)
{
    (void)n_in; (void)out_size; (void)ws_size;

    const int*   rows = (const int*)  d_in[0];   // adj_rows
    const int*   cols = (const int*)  d_in[1];   // adj_cols
    const float* vals = (const float*)d_in[2];   // adj_vals
    const float* u    = (const float*)d_in[3];   // uEmbeds [N_USERS*64]

    float* out = (float*)d_out;

    const int E  = in_sizes[0];                  // 1,600,000
    const int ND = in_sizes[3];                  // 100,000 * 64
    const int n4 = ND / 4;

    // Workspace: two fp32 ping-pong buffers of ND elements (2 x 25.6 MB).
    float* buf0 = (float*)d_ws;
    float* buf1 = buf0 + ND;

    const dim3 eB(256), eG((n4 + 255) / 256);
    const dim3 sB(256), sG((E + LGCN_TILE - 1) / LGCN_TILE);

    // out = u ; buf0 = 0
    lgcn_init<<<eG, eB, 0, stream>>>((float4*)out, (const float4*)u,
                                     (float4*)buf0, n4);
    // layer 1: buf0 = A*u ; out += buf0 ; buf1 = 0
    lgcn_scatter<<<sG, sB, 0, stream>>>(rows, cols, vals, u, buf0, E);
    lgcn_add_zero<<<eG, eB, 0, stream>>>((float4*)out, (const float4*)buf0,
                                         (float4*)buf1, n4);
    // layer 2: buf1 = A*buf0 ; out += buf1 ; buf0 = 0
    lgcn_scatter<<<sG, sB, 0, stream>>>(rows, cols, vals, buf0, buf1, E);
    lgcn_add_zero<<<eG, eB, 0, stream>>>((float4*)out, (const float4*)buf1,
                                         (float4*)buf0, n4);
    // layer 3: buf0 = A*buf1 ; out += buf0
    lgcn_scatter<<<sG, sB, 0, stream>>>(rows, cols, vals, buf1, buf0, E);
    lgcn_add_zero<<<eG, eB, 0, stream>>>((float4*)out, (const float4*)buf0,
                                         (float4*)nullptr, n4);
}